// Block_42408507081002
// MI455X (gfx1250) — compile-verified
//
#include <hip/hip_runtime.h>

#define BB   2
#define NN   196608
#define FF   64
#define KCH  8          // Chebyshev order
#define EE   (NN * 8)   // edges
#define EPSV 1e-5f
#define NEG_SLOPE 0.1f

typedef float v2f __attribute__((ext_vector_type(2)));
typedef float v8f __attribute__((ext_vector_type(8)));

// ---------------------------------------------------------------------------
// gfx1250 async global->LDS copy (ASYNCcnt path). LDS dest address is the low
// 32 bits of the generic shared pointer (LDS aperture offset).
// ---------------------------------------------------------------------------
__device__ __forceinline__ void async_load_b128_to_lds(unsigned lds_addr,
                                                       unsigned long long gaddr) {
    asm volatile("global_load_async_to_lds_b128 %0, %1, off"
                 :: "v"(lds_addr), "v"(gaddr) : "memory");
}
__device__ __forceinline__ void wait_asynccnt0() {
    asm volatile("s_wait_asynccnt 0" ::: "memory");
}

// ---------------------------------------------------------------------------
// CSR row pointers from sorted edge_rows via binary search (lower_bound).
// ---------------------------------------------------------------------------
__global__ void build_row_ptr(const int* __restrict__ rows, int* __restrict__ row_ptr) {
    int i = blockIdx.x * blockDim.x + threadIdx.x;
    if (i > NN) return;
    int lo = 0, hi = EE;
    while (lo < hi) {
        int mid = (lo + hi) >> 1;
        if (rows[mid] < i) lo = mid + 1; else hi = mid;
    }
    row_ptr[i] = lo;
}

// ---------------------------------------------------------------------------
// WMMA core shared by both kernels.
// A frag (16x4 f32): lane L -> M=L&15, VGPR v -> K = 2*(L>>4)+v
// B frag (4x16 f32): lane L -> N=L&15, VGPR v -> K = 2*(L>>4)+v
// C frag (16x16 f32): VGPR v -> M = v + 8*(L>>4), N = L&15
// ---------------------------------------------------------------------------
__device__ __forceinline__ void wmma_tile_64x64(const float* sXrow, const float* sW,
                                                int half, int l15, v8f c[4]) {
#pragma unroll
    for (int kk = 0; kk < 64; kk += 4) {
        v2f a;
        a.x = sXrow[l15 * 65 + kk + 2 * half];
        a.y = sXrow[l15 * 65 + kk + 2 * half + 1];
#pragma unroll
        for (int g = 0; g < 4; ++g) {
            v2f bf;
            bf.x = sW[(kk + 2 * half)     * 64 + g * 16 + l15];
            bf.y = sW[(kk + 2 * half + 1) * 64 + g * 16 + l15];
            c[g] = __builtin_amdgcn_wmma_f32_16x16x4_f32(
                false, a, false, bf, (short)0, c[g], false, false);
        }
    }
}

// ---------------------------------------------------------------------------
// k=0 only: acc = x @ W0 + bias (X read from global, W async-staged to LDS).
// ---------------------------------------------------------------------------
#define GW 4  // waves per block
__global__ void __launch_bounds__(GW * 32)
gemm_init_kernel(const float* __restrict__ X, const float* __restrict__ Wk,
                 const float* __restrict__ bias, float* __restrict__ acc) {
    __shared__ float sW[64 * 64];
    __shared__ float sX[GW][16 * 65];
    const int tid  = threadIdx.x;
    const int wave = tid >> 5;
    const int lane = tid & 31;
    const int half = lane >> 4;
    const int l15  = lane & 15;
    const int tileRow = (blockIdx.x * GW + wave) * 16;

    {   // async-stage Wk (16 KB): 1024 x B128, 8 per thread
        unsigned lds0 = (unsigned)(size_t)(&sW[0]);
        unsigned long long g0 = (unsigned long long)(size_t)Wk;
        for (int i = tid; i < 1024; i += GW * 32)
            async_load_b128_to_lds(lds0 + (unsigned)i * 16u, g0 + (unsigned long long)i * 16u);
    }
    for (int i = lane; i < 16 * 16; i += 32) {   // 16 rows x 16 float4
        int r = i >> 4, cc = (i & 15) * 4;
        float4 xv = *(const float4*)(X + (size_t)(tileRow + r) * FF + cc);
        float* p = &sX[wave][r * 65 + cc];
        p[0] = xv.x; p[1] = xv.y; p[2] = xv.z; p[3] = xv.w;
    }
    wait_asynccnt0();
    __syncthreads();

    v8f c[4];
#pragma unroll
    for (int g = 0; g < 4; ++g) {
        float bv = bias[g * 16 + l15];
#pragma unroll
        for (int v = 0; v < 8; ++v) c[g][v] = bv;
    }
    wmma_tile_64x64(sX[wave], sW, half, l15, c);
#pragma unroll
    for (int g = 0; g < 4; ++g)
#pragma unroll
        for (int v = 0; v < 8; ++v)
            acc[(size_t)(tileRow + v + 8 * half) * FF + g * 16 + l15] = c[g][v];
}

// ---------------------------------------------------------------------------
// Fused Chebyshev level: per wave, 16 rows of
//   xk[row,:] = alpha * sum_e vals[e]*src[b,cols[e],:] + beta * prev[row,:]
// written once to LDS + dst, then acc[tile] += xk_tile @ Wk via WMMA.
// prev may alias dst (elementwise; no __restrict__ on those).
// ---------------------------------------------------------------------------
#define FW 4  // waves per block
__global__ void __launch_bounds__(FW * 32)
cheb_fused_kernel(const float* __restrict__ src, const float* prev, float* dst,
                  const float* __restrict__ Wk, float* __restrict__ acc,
                  const int* __restrict__ cols, const float* __restrict__ vals,
                  const int* __restrict__ row_ptr, float alpha, float beta) {
    __shared__ float sW[64 * 64];
    __shared__ float sX[FW][16 * 65];
    const int tid  = threadIdx.x;
    const int wave = tid >> 5;
    const int lane = tid & 31;
    const int half = lane >> 4;
    const int l15  = lane & 15;
    const int tileRow = (blockIdx.x * FW + wave) * 16;     // global row in [0, B*N)
    const int b  = (tileRow >= NN) ? 1 : 0;                // N%16==0: no straddle
    const int n0 = tileRow - b * NN;
    const float* srcb = src + (size_t)b * NN * FF;

    {   // async-stage Wk (16 KB) into LDS
        unsigned lds0 = (unsigned)(size_t)(&sW[0]);
        unsigned long long g0 = (unsigned long long)(size_t)Wk;
        for (int i = tid; i < 1024; i += FW * 32)
            async_load_b128_to_lds(lds0 + (unsigned)i * 16u, g0 + (unsigned long long)i * 16u);
    }

    // SpMM for this wave's 16 rows: 2 channels per lane.
    for (int r = 0; r < 16; ++r) {
        int n  = n0 + r;
        int e0 = row_ptr[n], e1 = row_ptr[n + 1];
        float2 a2 = make_float2(0.f, 0.f);
        for (int e = e0; e < e1; e += 32) {
            int m = e1 - e; if (m > 32) m = 32;
            int   mycol = 0; float myval = 0.f;
            if (lane < m) {
                mycol = cols[e + lane];
                myval = vals[e + lane];
                // each lane prefetches a distinct upcoming gather row
                __builtin_prefetch(srcb + (size_t)mycol * FF, 0, 0);
            }
            for (int j = 0; j < m; ++j) {
                int   col = __shfl(mycol, j, 32);
                float val = __shfl(myval, j, 32);
                float2 xv = *(const float2*)(srcb + (size_t)col * FF + 2 * lane);
                a2.x += val * xv.x;
                a2.y += val * xv.y;
            }
        }
        size_t o  = (size_t)(tileRow + r) * FF + 2 * lane;
        float2 pv = *(const float2*)(prev + o);
        float2 rr;
        rr.x = alpha * a2.x + beta * pv.x;
        rr.y = alpha * a2.y + beta * pv.y;
        *(float2*)(dst + o) = rr;                  // x_k for next level
        float* p = &sX[wave][r * 65 + 2 * lane];   // A-tile for WMMA
        p[0] = rr.x; p[1] = rr.y;
    }
    wait_asynccnt0();
    __syncthreads();

    v8f c[4];
#pragma unroll
    for (int g = 0; g < 4; ++g)
#pragma unroll
        for (int v = 0; v < 8; ++v)
            c[g][v] = acc[(size_t)(tileRow + v + 8 * half) * FF + g * 16 + l15];
    wmma_tile_64x64(sX[wave], sW, half, l15, c);
#pragma unroll
    for (int g = 0; g < 4; ++g)
#pragma unroll
        for (int v = 0; v < 8; ++v)
            acc[(size_t)(tileRow + v + 8 * half) * FF + g * 16 + l15] = c[g][v];
}

// ---------------------------------------------------------------------------
// GroupNorm stage 1: per (b,g) partial sum / sumsq over strided rows.
// grid = (64, 16); block = 256.
// ---------------------------------------------------------------------------
__global__ void gn_reduce1(const float* __restrict__ acc, float2* __restrict__ partials) {
    int bg = blockIdx.y;            // b*8+g
    int b = bg >> 3, g = bg & 7;
    int j    = threadIdx.x & 7;
    int nOff = threadIdx.x >> 3;    // 0..31
    float s = 0.f, ss = 0.f;
    for (int n = blockIdx.x * 32 + nOff; n < NN; n += 64 * 32) {
        float v = acc[((size_t)b * NN + n) * FF + g * 8 + j];
        s += v; ss += v * v;
    }
    __shared__ float rs[256], rss[256];
    rs[threadIdx.x] = s; rss[threadIdx.x] = ss;
    __syncthreads();
    for (int off = 128; off > 0; off >>= 1) {
        if (threadIdx.x < off) {
            rs[threadIdx.x]  += rs[threadIdx.x + off];
            rss[threadIdx.x] += rss[threadIdx.x + off];
        }
        __syncthreads();
    }
    if (threadIdx.x == 0) partials[bg * 64 + blockIdx.x] = make_float2(rs[0], rss[0]);
}

// Stage 2: 16 wave32's, one per (b,g); finish with wave shuffles.
__global__ void gn_reduce2(const float2* __restrict__ partials, float2* __restrict__ stats) {
    int bg   = threadIdx.x >> 5;
    int lane = threadIdx.x & 31;
    float2 p0 = partials[bg * 64 + lane];
    float2 p1 = partials[bg * 64 + 32 + lane];
    float s = p0.x + p1.x, ss = p0.y + p1.y;
    for (int off = 16; off > 0; off >>= 1) {
        s  += __shfl_down(s,  off, 32);
        ss += __shfl_down(ss, off, 32);
    }
    if (lane == 0) {
        const float cnt = (float)NN * 8.0f;
        float mean = s / cnt;
        float var  = ss / cnt - mean * mean;
        stats[bg] = make_float2(mean, rsqrtf(var + EPSV));
    }
}

// Normalize + affine + LeakyReLU, float4-vectorized (4 channels share a group).
__global__ void gn_apply(float* __restrict__ y, const float2* __restrict__ stats,
                         const float* __restrict__ gamma, const float* __restrict__ beta) {
    size_t i4 = (size_t)blockIdx.x * blockDim.x + threadIdx.x;
    const size_t total4 = (size_t)BB * NN * (FF / 4);
    if (i4 >= total4) return;
    int    c4  = (int)(i4 & 15) * 4;
    size_t row = i4 >> 4;               // b*N + n
    int    b   = (int)(row / NN);
    int    g   = c4 >> 3;
    float2 st  = stats[b * 8 + g];
    float4 v   = *(float4*)(y + i4 * 4);
    float  vv[4] = {v.x, v.y, v.z, v.w}, o[4];
#pragma unroll
    for (int t = 0; t < 4; ++t) {
        float yn = (vv[t] - st.x) * st.y * gamma[c4 + t] + beta[c4 + t];
        o[t] = (yn >= 0.f) ? yn : NEG_SLOPE * yn;
    }
    *(float4*)(y + i4 * 4) = make_float4(o[0], o[1], o[2], o[3]);
}

// ---------------------------------------------------------------------------
extern "C" void kernel_launch(void* const* d_in, const int* in_sizes, int n_in,
                              void* d_out, int out_size, void* d_ws, size_t ws_size,
                              hipStream_t stream) {
    const float* x      = (const float*)d_in[0];
    const int*   erows  = (const int*)  d_in[1];
    const int*   ecols  = (const int*)  d_in[2];
    const float* evals  = (const float*)d_in[3];
    const float* weight = (const float*)d_in[4];   // [K,64,64]
    const float* bias   = (const float*)d_in[5];
    const float* gamma  = (const float*)d_in[6];
    const float* beta   = (const float*)d_in[7];
    float* acc = (float*)d_out;                    // [B,N,64] accumulator / output

    char* ws = (char*)d_ws;
    const size_t SZ = (size_t)BB * NN * FF * sizeof(float);   // 100.7 MB
    float* xA      = (float*)ws;
    float* xB      = (float*)(ws + SZ);
    int*   row_ptr = (int*)(ws + 2 * SZ);
    size_t rpBytes = (((size_t)(NN + 1) * 4 + 255) & ~(size_t)255);
    float2* partials = (float2*)(ws + 2 * SZ + rpBytes);
    float2* stats    = partials + 16 * 64;

    build_row_ptr<<<(NN + 256) / 256, 256, 0, stream>>>(erows, row_ptr);

    const int tiles = BB * NN / 16;   // 24576

    // k = 0: acc = x @ W0 + bias
    gemm_init_kernel<<<tiles / GW, GW * 32, 0, stream>>>(x, weight, bias, acc);

    // k = 1: xA = L x ; acc += xA @ W1   (alpha=1, beta=0)
    cheb_fused_kernel<<<tiles / FW, FW * 32, 0, stream>>>(
        x, x, xA, weight + 4096, acc, ecols, evals, row_ptr, 1.f, 0.f);

    // k = 2..7: x_k = 2 L x_{k-1} - x_{k-2} (ping-pong), acc += x_k @ Wk
    const float* prevp = x;
    const float* curp  = xA;
    float* dsts[2] = {xB, xA};
    for (int k = 2; k < KCH; ++k) {
        float* dst = dsts[k & 1];
        cheb_fused_kernel<<<tiles / FW, FW * 32, 0, stream>>>(
            curp, prevp, dst, weight + (size_t)k * 4096, acc,
            ecols, evals, row_ptr, 2.f, -1.f);
        prevp = curp; curp = dst;
    }

    dim3 g1(64, 16);
    gn_reduce1<<<g1, 256, 0, stream>>>(acc, partials);
    gn_reduce2<<<1, 512, 0, stream>>>(partials, stats);
    size_t total4 = (size_t)BB * NN * (FF / 4);
    gn_apply<<<(int)((total4 + 255) / 256), 256, 0, stream>>>(acc, stats, gamma, beta);
}